// Preprocessor_75763223101807
// MI455X (gfx1250) — compile-verified
//
#include <hip/hip_runtime.h>

// CDNA5 / gfx1250, wave32. One wave per pixel.
typedef __attribute__((ext_vector_type(2))) float v2f;
typedef __attribute__((ext_vector_type(8))) float v8f;

#define LOG127_5 4.8481166f   // ln(127.5)

// Native-pipe transcendentals (v_exp_f32 / v_log_f32 / v_rcp_f32).
__device__ __forceinline__ float fast_exp(float v) {
    return __builtin_amdgcn_exp2f(v * 1.4426950408889634f);  // exp2(v*log2e)
}
__device__ __forceinline__ float fast_log(float v) {
    return __builtin_amdgcn_logf(v) * 0.6931471805599453f;   // log2(v)*ln2
}
__device__ __forceinline__ float sp_softplus(float v) {
    // jax.nn.softplus: max(v,0) + log1p(exp(-|v|)) (stable form, native pipes)
    return fmaxf(v, 0.0f) + fast_log(1.0f + fast_exp(-fabsf(v)));
}
__device__ __forceinline__ float sp_sigmoid(float v) {
    return __builtin_amdgcn_rcpf(1.0f + fast_exp(-v));
}

// x: [B,3,H,W], l: [B,100,H,W], out: [B,3,H,W]
__global__ __launch_bounds__(256) void mol_decode_kernel(
    const float* __restrict__ x, const float* __restrict__ l,
    const int* __restrict__ epsp, float* __restrict__ out,
    int npix, int HW)
{
    __shared__ float sbuf[8][104];   // per-wave pixel params: l[0..99], x[100..102]
    const int lane = threadIdx.x & 31;
    const int wv   = threadIdx.x >> 5;
    int pix = blockIdx.x * 8 + wv;
    if (pix >= npix) pix = npix - 1;           // redundant recompute, deterministic
    const int b  = pix / HW;
    const int hw = pix - b * HW;

    const float* lsrc = l + (size_t)b * 100 * HW + hw;
    __builtin_prefetch(lsrc + (size_t)lane * HW, 0, 0);   // global_prefetch_b8
    for (int j = lane; j < 100; j += 32)
        sbuf[wv][j] = lsrc[(size_t)j * HW];
    if (lane < 3)
        sbuf[wv][100 + lane] = x[((size_t)b * 3 + lane) * HW + hw];
    __syncthreads();

    const float* P = sbuf[wv];
    const int eps = epsp[0];

    // log-softmax denominator over logit_probs P[0..9]
    float mmax = -INFINITY;
    #pragma unroll
    for (int k = 0; k < 10; ++k) mmax = fmaxf(mmax, P[k]);
    float ssum = 0.0f;
    #pragma unroll
    for (int k = 0; k < 10; ++k) ssum += fast_exp(P[k] - mmax);
    const float lse = mmax + fast_log(ssum);

    const int half = lane >> 4;     // which 16-lane half (mixture block / B K-block)
    const int ncol = lane & 15;     // D column within tile

    float xo0 = 0.0f, xo1 = 0.0f;   // sequential channel outputs

    for (int c = 0; c < 3; ++c) {
        // box-constraint window from ORIGINAL x (penalty)
        const float xin = P[100 + c];
        const int recov = (int)__fadd_rn(__fmul_rn(xin, 127.5f), 127.5f);
        const int lo = max(recov - eps, 0);
        const int hi = min(recov + eps, 255);

        // A-matrix: row k = [inv_k, -inv_k*mean_k, 0, 0]; lanes 0-15 hold K0/K1
        float a_inv = 0.0f, a_bias = 0.0f;
        if (lane < 10) {
            float mean = P[10 + c * 30 + lane];
            if (c == 1) mean += tanhf(P[30 + lane]) * xo0;
            if (c == 2) mean += tanhf(P[60 + lane]) * xo0 + tanhf(P[90 + lane]) * xo1;
            const float lsc = fmaxf(P[20 + c * 30 + lane], -7.0f);
            const float iv  = fast_exp(-lsc);
            a_inv  = iv;
            a_bias = -iv * mean;
        }
        v2f A; A.x = a_inv; A.y = a_bias;

        float bestv = -INFINITY;
        int   besti = lo;

        for (int t0 = lo; t0 <= hi; t0 += 16) {
            const int t = t0 + ncol;            // color bin for this lane's column
            // B (4x16): row0 = threshold colors, row1 = ones, rows 2,3 = 0.
            // Low lanes carry K0 (in .x) and K1 (in .y); high lanes K2/K3 = 0.
            const float fb = half ? 0.0f : 1.0f;
            v2f BL, BR, BM;
            BL.x = half ? 0.0f : (float)(t - 128) * (1.0f / 128.0f);    BL.y = fb; // colors[t]
            BR.x = half ? 0.0f : (float)(t - 127) * (1.0f / 128.0f);    BR.y = fb; // colors[t+1]
            BM.x = half ? 0.0f : ((float)t - 127.5f) * (1.0f / 128.0f); BM.y = fb; // bin midpoint
            const v8f zero = {};
            // arg[k][t] = inv_k*color_t - inv_k*mean_k   (three threshold sets)
            v8f DL = __builtin_amdgcn_wmma_f32_16x16x4_f32(false, A, false, BL, (short)0, zero, false, false);
            v8f DR = __builtin_amdgcn_wmma_f32_16x16x4_f32(false, A, false, BR, (short)0, zero, false, false);
            v8f DM = __builtin_amdgcn_wmma_f32_16x16x4_f32(false, A, false, BM, (short)0, zero, false, false);

            // online logsumexp over this lane's 8 mixture slots (k = half*8 + v)
            float mx = -INFINITY, sm = 0.0f;
            #pragma unroll
            for (int v = 0; v < 8; ++v) {
                const int kk = half * 8 + v;
                const float aL = DL[v], aR = DR[v], aM = DM[v];
                float lp;
                if (t == 0) {
                    lp = aR - sp_softplus(aR);              // log CDF at left edge
                } else if (t == 255) {
                    lp = -sp_softplus(aL);                  // log(1 - CDF) at right edge
                } else {
                    const float pdf = sp_sigmoid(aR) - sp_sigmoid(aL);
                    if (pdf > 1e-5f) {
                        lp = fast_log(fmaxf(pdf, 1e-12f));
                    } else {
                        const float lsc = fmaxf(P[20 + c * 30 + kk], -7.0f);
                        lp = aM - lsc - 2.0f * sp_softplus(aM) - LOG127_5;
                    }
                }
                lp += P[kk] - lse;                          // + log_softmax weight
                if (kk < 10) {
                    if (lp > mx) { sm = sm * fast_exp(mx - lp) + 1.0f; mx = lp; }
                    else         { sm += fast_exp(lp - mx); }
                }
            }
            // merge the two mixture halves (lane L <-> L^16 hold same column t)
            const float om = __shfl_xor(mx, 16, 32);
            const float os = __shfl_xor(sm, 16, 32);
            const float M  = fmaxf(mx, om);
            const float Lt = M + fast_log(sm * fast_exp(mx - M) + os * fast_exp(om - M));
            // per-lane running argmax; t strictly increases -> '>' keeps first max
            if (t <= hi && Lt > bestv) { bestv = Lt; besti = t; }
        }

        // wave-wide argmax butterfly; ties -> lower index (jnp.argmax semantics)
        #pragma unroll
        for (int off = 16; off > 0; off >>= 1) {
            const float ov = __shfl_xor(bestv, off, 32);
            const int   oi = __shfl_xor(besti, off, 32);
            if (ov > bestv || (ov == bestv && oi < besti)) { bestv = ov; besti = oi; }
        }
        const float xc = ((float)besti - 127.5f) / 127.5f;
        if (c == 0) xo0 = xc;
        else if (c == 1) xo1 = xc;
        if (lane == 0) out[((size_t)b * 3 + c) * HW + hw] = xc;
    }
}

extern "C" void kernel_launch(void* const* d_in, const int* in_sizes, int n_in,
                              void* d_out, int out_size, void* d_ws, size_t ws_size,
                              hipStream_t stream) {
    const float* x   = (const float*)d_in[0];   // [16,3,32,32]
    const float* l   = (const float*)d_in[1];   // [16,100,32,32]
    const int*   eps = (const int*)d_in[2];     // scalar
    float* out = (float*)d_out;                 // [16,3,32,32]
    const int npix = in_sizes[0] / 3;           // B*H*W = 16384
    const int HW   = 1024;                      // 32*32 (fixed problem shape)
    const int blocks = (npix + 7) / 8;          // 8 pixels (waves) per 256-thread block
    mol_decode_kernel<<<blocks, 256, 0, stream>>>(x, l, eps, out, npix, HW);
}